// SegSparseVoxHead_9199819948458
// MI455X (gfx1250) — compile-verified
//
#include <hip/hip_runtime.h>
#include <hip/hip_bf16.h>
#include <cstdint>
#include <cstddef>

// ---------------------------------------------------------------------------
// SegSparseVoxHead for MI455X (gfx1250, wave32)
//   Phase A: dropout + segment_max via counting-sort (no float atomics)
//   Phase B: MLP head with V_WMMA_F32_16X16X4_F32 (exact fp32 matmul)
// ---------------------------------------------------------------------------

typedef __attribute__((ext_vector_type(2))) float v2f;
typedef __attribute__((ext_vector_type(8))) float v8f;

constexpr int   kSegs  = 50000;
constexpr int   kCIn   = 96;
constexpr int   kCHid  = 256;
constexpr int   kCOut  = 128;
constexpr float kPDrop = 0.4f;
constexpr float kScale = 1.0f / (1.0f - kPDrop);

__device__ __forceinline__ v8f wmma_f32_16x16x4(v2f a, v2f b, v8f c) {
#if defined(__HIP_DEVICE_COMPILE__)
  // 8 args: (neg_a, A, neg_b, B, c_mod, C, reuse_a, reuse_b)
  return __builtin_amdgcn_wmma_f32_16x16x4_f32(false, a, false, b, (short)0, c,
                                               false, false);
#else
  (void)a; (void)b;
  return c;
#endif
}

// --------------------------- Phase A kernels -------------------------------

__global__ void zero_counts_kernel(int* __restrict__ counts) {
  int i = blockIdx.x * blockDim.x + threadIdx.x;
  if (i < kSegs) counts[i] = 0;
}

__global__ void count_kernel(const int* __restrict__ cid,
                             int* __restrict__ counts, int n) {
  int i = blockIdx.x * blockDim.x + threadIdx.x;
  if (i < n) atomicAdd(&counts[cid[i]], 1);
}

// Single-workgroup exclusive scan over 50000 bins (1024 threads, 49 bins each).
__global__ __launch_bounds__(1024)
void scan_kernel(const int* __restrict__ counts, int* __restrict__ offsets,
                 int* __restrict__ cursor) {
  __shared__ int sums[1024];
  const int t = threadIdx.x;
  const int CHUNK = (kSegs + 1023) / 1024;  // 49
  int s0 = t * CHUNK;
  int s1 = s0 + CHUNK; if (s1 > kSegs) s1 = kSegs;
  int acc = 0;
  for (int i = s0; i < s1; ++i) acc += counts[i];
  sums[t] = acc;
  __syncthreads();
  // Hillis-Steele inclusive scan across the 1024 chunk sums.
  for (int d = 1; d < 1024; d <<= 1) {
    int v = (t >= d) ? sums[t - d] : 0;
    __syncthreads();
    sums[t] += v;
    __syncthreads();
  }
  int run = (t == 0) ? 0 : sums[t - 1];
  for (int i = s0; i < s1; ++i) {
    offsets[i] = run;
    cursor[i]  = run;
    run += counts[i];
  }
}

__global__ void scatter_kernel(const int* __restrict__ cid,
                               int* __restrict__ cursor,
                               int* __restrict__ pidx, int n) {
  int i = blockIdx.x * blockDim.x + threadIdx.x;
  if (i < n) {
    int p = atomicAdd(&cursor[cid[i]], 1);
    pidx[p] = i;
  }
}

// One wave per segment; lane L owns channels {L, L+32, L+64}.
// Each point row is 96 f32 = 384 B contiguous -> three coalesced 128 B loads.
__global__ __launch_bounds__(256)
void segmax_kernel(const float* __restrict__ feats,
                   const float* __restrict__ mask,
                   const int* __restrict__ pidx,
                   const int* __restrict__ offsets,
                   const int* __restrict__ counts,
                   float* __restrict__ pooled) {
  const int wave = threadIdx.x >> 5;
  const int lane = threadIdx.x & 31;
  const int s = blockIdx.x * 8 + wave;
  if (s >= kSegs) return;  // wave-uniform
  const int start = offsets[s];
  const int cnt   = counts[s];
  float m0 = -INFINITY, m1 = -INFINITY, m2 = -INFINITY;
  for (int j = 0; j < cnt; ++j) {
    const int p = pidx[start + j];
    const size_t base = (size_t)p * kCIn;
    float f0 = feats[base + lane];
    float f1 = feats[base + lane + 32];
    float f2 = feats[base + lane + 64];
    float q0 = mask[base + lane];
    float q1 = mask[base + lane + 32];
    float q2 = mask[base + lane + 64];
    float x0 = (q0 >= kPDrop) ? f0 * kScale : 0.0f;
    float x1 = (q1 >= kPDrop) ? f1 * kScale : 0.0f;
    float x2 = (q2 >= kPDrop) ? f2 * kScale : 0.0f;
    m0 = fmaxf(m0, x0);
    m1 = fmaxf(m1, x1);
    m2 = fmaxf(m2, x2);
  }
  const size_t ob = (size_t)s * kCIn;
  // Empty segment -> segment_max is -inf -> reference maps non-finite to 0.
  pooled[ob + lane]      = cnt ? m0 : 0.0f;
  pooled[ob + lane + 32] = cnt ? m1 : 0.0f;
  pooled[ob + lane + 64] = cnt ? m2 : 0.0f;
}

// --------------------------- Phase B: WMMA MLP -----------------------------
// Block = 256 threads = 8 waves, handles 16 segment rows.
// Layer 1: wave w computes h-tiles n=[32w,32w+32) -> relu -> LDS.
// Layer 2: wave w computes out-tile n=[16w,16w+16) reading A from LDS.
// f32 WMMA layouts (wave32): half = lane>=16, lid = lane&15
//   A 16x4 : a0 = A[lid][k0+2*half],   a1 = A[lid][k0+2*half+1]
//   B 4x16 : b0 = B[k0+2*half][n0+lid], b1 = B[k0+2*half+1][n0+lid]
//   C/D    : c[r] = D[r + 8*half][n0+lid]
__global__ __launch_bounds__(256)
void mlp_kernel(const float* __restrict__ pooled,
                const float* __restrict__ W1, const float* __restrict__ b1,
                const float* __restrict__ W2, const float* __restrict__ b2,
                float* __restrict__ out) {
  constexpr int HSTR = kCHid + 1;  // 257: break LDS bank alignment
  __shared__ float Hs[16 * HSTR];

  const int wave = threadIdx.x >> 5;
  const int lane = threadIdx.x & 31;
  const int half = lane >> 4;
  const int lid  = lane & 15;
  const int kh   = 2 * half;
  const int seg0 = blockIdx.x * 16;  // 50000 == 3125 * 16, exact

  // ---- Layer 1: h = relu(pooled @ W1 + b1), staged in LDS ----
  const float* Arow = pooled + (size_t)(seg0 + lid) * kCIn;
  for (int nt = wave * 2; nt < wave * 2 + 2; ++nt) {
    const int n0 = nt * 16;
    v8f acc = {0.f, 0.f, 0.f, 0.f, 0.f, 0.f, 0.f, 0.f};
#pragma unroll 4
    for (int k0 = 0; k0 < kCIn; k0 += 4) {
      v2f a = {Arow[k0 + kh], Arow[k0 + kh + 1]};
      v2f b = {W1[(size_t)(k0 + kh) * kCHid + n0 + lid],
               W1[(size_t)(k0 + kh + 1) * kCHid + n0 + lid]};
      acc = wmma_f32_16x16x4(a, b, acc);
    }
    const float bias = b1[n0 + lid];
#pragma unroll
    for (int r = 0; r < 8; ++r) {
      float h = acc[r] + bias;
      h = h > 0.f ? h : 0.f;
      Hs[(r + 8 * half) * HSTR + n0 + lid] = h;
    }
  }
  __syncthreads();

  // ---- Layer 2: out = h @ W2 + b2 ----
  {
    const int n0 = wave * 16;
    v8f acc = {0.f, 0.f, 0.f, 0.f, 0.f, 0.f, 0.f, 0.f};
#pragma unroll 4
    for (int k0 = 0; k0 < kCHid; k0 += 4) {
      v2f a = {Hs[lid * HSTR + k0 + kh], Hs[lid * HSTR + k0 + kh + 1]};
      v2f b = {W2[(size_t)(k0 + kh) * kCOut + n0 + lid],
               W2[(size_t)(k0 + kh + 1) * kCOut + n0 + lid]};
      acc = wmma_f32_16x16x4(a, b, acc);
    }
    const float bias = b2[n0 + lid];
#pragma unroll
    for (int r = 0; r < 8; ++r) {
      out[(size_t)(seg0 + r + 8 * half) * kCOut + n0 + lid] = acc[r] + bias;
    }
  }
}

// ------------------------------- Launcher ----------------------------------

extern "C" void kernel_launch(void* const* d_in, const int* in_sizes, int n_in,
                              void* d_out, int out_size, void* d_ws,
                              size_t ws_size, hipStream_t stream) {
  (void)n_in; (void)out_size; (void)ws_size;

  const float* feats = (const float*)d_in[0];
  const int*   cid   = (const int*)d_in[1];
  const float* mask  = (const float*)d_in[2];
  const float* W1    = (const float*)d_in[3];
  const float* b1    = (const float*)d_in[4];
  const float* W2    = (const float*)d_in[5];
  const float* b2    = (const float*)d_in[6];
  float*       out   = (float*)d_out;
  const int    N     = in_sizes[1];  // number of points (cluster_ids)

  // Workspace carve-out (256 B aligned): counts, offsets, cursor, pidx, pooled
  uint8_t* ws = (uint8_t*)d_ws;
  size_t off = 0;
  auto carve = [&](size_t bytes) -> void* {
    off = (off + 255) & ~(size_t)255;
    void* p = ws + off;
    off += bytes;
    return p;
  };
  int*   counts  = (int*)carve((size_t)kSegs * sizeof(int));
  int*   offsets = (int*)carve((size_t)kSegs * sizeof(int));
  int*   cursor  = (int*)carve((size_t)kSegs * sizeof(int));
  int*   pidx    = (int*)carve((size_t)N * sizeof(int));
  float* pooled  = (float*)carve((size_t)kSegs * kCIn * sizeof(float));

  zero_counts_kernel<<<(kSegs + 255) / 256, 256, 0, stream>>>(counts);
  count_kernel<<<(N + 255) / 256, 256, 0, stream>>>(cid, counts, N);
  scan_kernel<<<1, 1024, 0, stream>>>(counts, offsets, cursor);
  scatter_kernel<<<(N + 255) / 256, 256, 0, stream>>>(cid, cursor, pidx, N);
  segmax_kernel<<<(kSegs + 7) / 8, 256, 0, stream>>>(feats, mask, pidx,
                                                     offsets, counts, pooled);
  mlp_kernel<<<kSegs / 16, 256, 0, stream>>>(pooled, W1, b1, W2, b2, out);
}